// AttentionDecoderLSTM_197568496130
// MI455X (gfx1250) — compile-verified
//
#include <hip/hip_runtime.h>
#include <hip/hip_bf16.h>

// CDNA5 / gfx1250: wave32, WMMA f32 16x16x4 for fp32 GEMV streaming.
typedef float v2f __attribute__((ext_vector_type(2)));
typedef float v8f __attribute__((ext_vector_type(8)));

#define WAVES_PER_BLOCK 4

__device__ __forceinline__ v8f wmma_f32_k4(v2f a, v2f b, v8f c) {
    return __builtin_amdgcn_wmma_f32_16x16x4_f32(false, a, false, b, (short)0, c,
                                                 false, false);
}

__device__ __forceinline__ float sigmoid_f(float x) {
    return 1.0f / (1.0f + __expf(-x));
}

// ---------------------------------------------------------------------------
// y[r] = dot(W[r, 0:K], v) + bias[r]   (row-major W, R rows; 16 rows per wave)
// A-matrix = 16 rows of W (lane<16: k0..k0+1, lane>=16: k0+2..k0+3)
// B-matrix = v replicated across all 16 columns -> every D column == y
// K-loop: outer step 64 (one WGP-scope prefetch), inner 16 unrolled WMMAs.
// ---------------------------------------------------------------------------
__global__ void gemv_rows_wmma(const float* __restrict__ W,
                               const float* __restrict__ v,
                               const float* __restrict__ bias,
                               float* __restrict__ y, int K) {
    __shared__ float sv[1024];
    const int tid = threadIdx.x;
    for (int i = tid; i < K; i += blockDim.x) sv[i] = v[i];
    __syncthreads();

    const int wave   = tid >> 5;
    const int lane   = tid & 31;
    const int laneLo = lane & 15;
    const int hi     = lane >> 4;                       // 0: k0..1, 1: k0+2..3
    const long row0  = (long)(blockIdx.x * WAVES_PER_BLOCK + wave) * 16;
    const float* wrow = W + (row0 + laneLo) * (long)K;

    v8f acc = {};
    for (int k0 = 0; k0 < K; k0 += 64) {
        __builtin_prefetch(wrow + k0 + 64, 0, 3);  // next 256B of this row
#pragma unroll
        for (int kk = 0; kk < 64; kk += 4) {
            const int ka = k0 + kk + 2 * hi;
            v2f a = *(const v2f*)(wrow + ka);
            v2f b; b.x = sv[ka]; b.y = sv[ka + 1];
            acc = wmma_f32_k4(a, b, acc);
        }
    }
    // D: lane holds column N=laneLo, rows M = g + 8*hi. Column 0 suffices.
    if (laneLo == 0) {
#pragma unroll
        for (int g = 0; g < 8; ++g) {
            const long r = row0 + g + 8 * hi;
            float val = acc[g];
            if (bias) val += bias[r];
            y[r] = val;
        }
    }
}

// ---------------------------------------------------------------------------
// ctx[j] = sum_i a[i] * E[i][j]   (E is S x K row-major; 16 columns per wave)
// A-matrix = a[i0..i0+3] replicated down the 16 rows
// B-matrix = E tile: VGPR0 -> row i0+2*hi, VGPR1 -> row i0+1+2*hi, N=laneLo
// ---------------------------------------------------------------------------
__global__ void gemv_cols_wmma(const float* __restrict__ E,
                               const float* __restrict__ a,
                               float* __restrict__ ctx, int S, int K) {
    __shared__ float sa[2048];
    const int tid = threadIdx.x;
    for (int i = tid; i < S; i += blockDim.x) sa[i] = a[i];
    __syncthreads();

    const int wave   = tid >> 5;
    const int lane   = tid & 31;
    const int laneLo = lane & 15;
    const int hi     = lane >> 4;
    const int j0     = (blockIdx.x * WAVES_PER_BLOCK + wave) * 16;

    v8f acc = {};
    for (int i0 = 0; i0 < S; i0 += 16) {
#pragma unroll
        for (int ii = 0; ii < 16; ii += 4) {
            const int ia = i0 + ii + 2 * hi;
            v2f av; av.x = sa[ia]; av.y = sa[ia + 1];
            v2f bv;
            bv.x = E[(long)ia * K + j0 + laneLo];
            bv.y = E[(long)(ia + 1) * K + j0 + laneLo];
            acc = wmma_f32_k4(av, bv, acc);
        }
    }
    // All D rows identical; lane's column is N=laneLo -> take M=0 (VGPR0).
    if (hi == 0) ctx[j0 + laneLo] = acc[0];
}

// ---------------------------------------------------------------------------
// softmax over 2048 scores, single block of 1024 threads (32 waves, wave32)
// ---------------------------------------------------------------------------
__global__ void softmax2048(const float* __restrict__ scores,
                            float* __restrict__ attn_ws,
                            float* __restrict__ attn_out) {
    __shared__ float red[32];
    __shared__ float s_max, s_sum;
    const int tid = threadIdx.x;
    float a = scores[tid], b = scores[tid + 1024];
    float m = fmaxf(a, b);
#pragma unroll
    for (int off = 16; off; off >>= 1) m = fmaxf(m, __shfl_xor(m, off, 32));
    if ((tid & 31) == 0) red[tid >> 5] = m;
    __syncthreads();
    if (tid < 32) {
        float mm = red[tid];
#pragma unroll
        for (int off = 16; off; off >>= 1) mm = fmaxf(mm, __shfl_xor(mm, off, 32));
        if (tid == 0) s_max = mm;
    }
    __syncthreads();
    float ea = __expf(a - s_max), eb = __expf(b - s_max);
    float s = ea + eb;
#pragma unroll
    for (int off = 16; off; off >>= 1) s += __shfl_xor(s, off, 32);
    if ((tid & 31) == 0) red[tid >> 5] = s;
    __syncthreads();
    if (tid < 32) {
        float ss = red[tid];
#pragma unroll
        for (int off = 16; off; off >>= 1) ss += __shfl_xor(ss, off, 32);
        if (tid == 0) s_sum = ss;
    }
    __syncthreads();
    const float inv = 1.0f / s_sum;
    attn_ws[tid] = ea * inv;  attn_ws[tid + 1024] = eb * inv;
    attn_out[tid] = ea * inv; attn_out[tid + 1024] = eb * inv;
}

// ---------------------------------------------------------------------------
// Fused LSTM cell: block handles 16 hidden indices j0..j0+15.
// Wave g computes gate-g rows via two accumulated WMMA GEMVs
// (W_ih . x  +  W_hh . hv), then 16 threads do the elementwise cell update.
// ---------------------------------------------------------------------------
__global__ void lstm_wmma(const float* __restrict__ Wih,
                          const float* __restrict__ Whh,
                          const float* __restrict__ bih,
                          const float* __restrict__ bhh,
                          const float* __restrict__ x_base,
                          const int* __restrict__ idx,  // optional row index
                          const float* __restrict__ hv,
                          const float* __restrict__ c_prev,
                          float* __restrict__ h_out,
                          float* __restrict__ c_out) {
    __shared__ float sx[1024];
    __shared__ float sh[1024];
    __shared__ float gates[4][16];
    const int tid = threadIdx.x;  // 128 threads = 4 waves
    const float* x = x_base + (idx ? (long)idx[0] * 1024 : 0);
    for (int i = tid; i < 1024; i += 128) { sx[i] = x[i]; sh[i] = hv[i]; }
    __syncthreads();

    const int wave   = tid >> 5;  // gate index: 0=i 1=f 2=g 3=o
    const int lane   = tid & 31;
    const int laneLo = lane & 15;
    const int hi     = lane >> 4;
    const int j0     = blockIdx.x * 16;
    const long row0  = (long)wave * 1024 + j0;
    const float* wi  = Wih + (row0 + laneLo) * 1024;
    const float* wh  = Whh + (row0 + laneLo) * 1024;

    v8f acc = {};
    for (int k0 = 0; k0 < 1024; k0 += 64) {
        __builtin_prefetch(wi + k0 + 64, 0, 3);
#pragma unroll
        for (int kk = 0; kk < 64; kk += 4) {
            const int ka = k0 + kk + 2 * hi;
            v2f a = *(const v2f*)(wi + ka);
            v2f b; b.x = sx[ka]; b.y = sx[ka + 1];
            acc = wmma_f32_k4(a, b, acc);
        }
    }
    for (int k0 = 0; k0 < 1024; k0 += 64) {
        __builtin_prefetch(wh + k0 + 64, 0, 3);
#pragma unroll
        for (int kk = 0; kk < 64; kk += 4) {
            const int ka = k0 + kk + 2 * hi;
            v2f a = *(const v2f*)(wh + ka);
            v2f b; b.x = sh[ka]; b.y = sh[ka + 1];
            acc = wmma_f32_k4(a, b, acc);
        }
    }
    if (laneLo == 0) {
#pragma unroll
        for (int g = 0; g < 8; ++g) {
            const int jj = g + 8 * hi;
            const long r = row0 + jj;
            gates[wave][jj] = acc[g] + bih[r] + bhh[r];
        }
    }
    __syncthreads();
    if (tid < 16) {
        const int j = j0 + tid;
        const float si = sigmoid_f(gates[0][tid]);
        const float sf = sigmoid_f(gates[1][tid]);
        const float tg = tanhf(gates[2][tid]);
        const float so = sigmoid_f(gates[3][tid]);
        const float cn = sf * c_prev[j] + si * tg;
        c_out[j] = cn;
        h_out[j] = so * tanhf(cn);
    }
}

// ---------------------------------------------------------------------------
extern "C" void kernel_launch(void* const* d_in, const int* in_sizes, int n_in,
                              void* d_out, int out_size, void* d_ws, size_t ws_size,
                              hipStream_t stream) {
    const int*   ids  = (const int*)d_in[0];
    const float* h    = (const float*)d_in[1];   // (2,1,1024)
    const float* c    = (const float*)d_in[2];   // (2,1,1024)
    const float* E    = (const float*)d_in[3];   // (2048,1024)
    const float* emb  = (const float*)d_in[4];   // (32000,1024)
    const float* Wih0 = (const float*)d_in[5];
    const float* Whh0 = (const float*)d_in[6];
    const float* bih0 = (const float*)d_in[7];
    const float* bhh0 = (const float*)d_in[8];
    const float* Wih1 = (const float*)d_in[9];
    const float* Whh1 = (const float*)d_in[10];
    const float* bih1 = (const float*)d_in[11];
    const float* bhh1 = (const float*)d_in[12];
    const float* Wout = (const float*)d_in[13];  // (32000,1024)
    const float* bout = (const float*)d_in[14];

    float* out = (float*)d_out;
    float* ws  = (float*)d_ws;

    // workspace layout (floats)
    float* scores = ws;          // 2048
    float* attn   = ws + 2048;   // 2048
    float* ctx    = ws + 4096;   // 1024

    // output layout: logits(32000) | h0(1024) h1(1024) | c0(1024) c1(1024) | attn(2048)
    float* logits   = out;
    float* h0       = out + 32000;
    float* h1       = out + 33024;
    float* c0       = out + 34048;
    float* c1       = out + 35072;
    float* attn_out = out + 36096;

    // 1) scores = E . h[0,0]                 (2048 rows / 64 rows per block)
    gemv_rows_wmma<<<32, 128, 0, stream>>>(E, h, nullptr, scores, 1024);
    // 2) attn = softmax(scores)
    softmax2048<<<1, 1024, 0, stream>>>(scores, attn, attn_out);
    // 3) ctx = attn . E                     (1024 cols / 64 cols per block)
    gemv_cols_wmma<<<16, 128, 0, stream>>>(E, attn, ctx, 2048, 1024);
    // 4) layer 0: x = embedding[ids[0]], h_prev = ctx
    lstm_wmma<<<64, 128, 0, stream>>>(Wih0, Whh0, bih0, bhh0,
                                      emb, ids, ctx, c, h0, c0);
    // 5) layer 1: x = h0, h_prev = ctx
    lstm_wmma<<<64, 128, 0, stream>>>(Wih1, Whh1, bih1, bhh1,
                                      h0, nullptr, ctx, c + 1024, h1, c1);
    // 6) logits = W_out . h1 + b_out        (32000 rows / 64 per block)
    gemv_rows_wmma<<<500, 128, 0, stream>>>(Wout, h1, bout, logits, 1024);
}